// RGCN_8778913153226
// MI455X (gfx1250) — compile-verified
//
#include <hip/hip_runtime.h>
#include <cstdint>

typedef float v2f __attribute__((ext_vector_type(2)));
typedef float v8f __attribute__((ext_vector_type(8)));
typedef int v4i __attribute__((vector_size(16)));

static constexpr int NN = 100000;   // nodes
static constexpr int DD = 128;      // hidden dim
static constexpr int EE = 1600000;  // edges
static constexpr int RR = 2;        // relations
static constexpr float BN_EPS = 1e-5f;

__device__ __forceinline__ v8f wmma_f32(v2f a, v2f b, v8f c) {
  // D = A(16x4 f32) * B(4x16 f32) + C(16x16 f32)
  return __builtin_amdgcn_wmma_f32_16x16x4_f32(false, a, false, b, (short)0, c,
                                               false, false);
}

__device__ __forceinline__ void atomAddF(float* p, float v) {
  __hip_atomic_fetch_add(p, v, __ATOMIC_RELAXED, __HIP_MEMORY_SCOPE_AGENT);
}

// ---- gfx1250 async global->LDS copy (ASYNCcnt path), guarded w/ fallback ----
#if defined(__has_builtin)
#if __has_builtin(__builtin_amdgcn_global_load_async_to_lds_b128)
#define HAS_ASYNC_LDS 1
#endif
#endif

typedef __attribute__((address_space(1))) v4i* as1_v4i_ptr;
typedef __attribute__((address_space(3))) v4i* as3_v4i_ptr;

__device__ __forceinline__ void async_copy16(void* lds, const void* g) {
#ifdef HAS_ASYNC_LDS
  __builtin_amdgcn_global_load_async_to_lds_b128(
      (as1_v4i_ptr)(const_cast<void*>(g)), (as3_v4i_ptr)(lds), 0, 0);
#else
  *(float4*)lds = *(const float4*)g;
#endif
}

__device__ __forceinline__ void async_wait0() {
#ifdef HAS_ASYNC_LDS
#if __has_builtin(__builtin_amdgcn_s_wait_asynccnt)
  __builtin_amdgcn_s_wait_asynccnt(0);
#else
  asm volatile("s_wait_asynccnt 0" ::: "memory");
#endif
#endif
}

// ---------------------------------------------------------------- degree count
__global__ void k_count(const int* __restrict__ ei, const int* __restrict__ et,
                        float* __restrict__ cnt) {
  int e = blockIdx.x * blockDim.x + threadIdx.x;
  if (e >= EE) return;
  int d = ei[EE + e];
  int r = et[e];
  atomAddF(&cnt[d * RR + r], 1.0f);
}

// -------------------------------------------------- edge aggregation (1 wave/edge)
// agg[r][dst] += h[src] / max(cnt[dst,r],1)   (mean aggregation, pre-transform)
__global__ void k_aggregate(const float* __restrict__ h, const int* __restrict__ ei,
                            const int* __restrict__ et, const float* __restrict__ cnt,
                            float* __restrict__ agg) {
  int gid = blockIdx.x * blockDim.x + threadIdx.x;
  int e = gid >> 5;          // wave32: one wave per edge
  int lane = gid & 31;       // lane handles 4 contiguous floats of D=128
  if (e >= EE) return;
  int s = ei[e];
  int d = ei[EE + e];
  int r = et[e];
  float sc = 1.0f / fmaxf(cnt[d * RR + r], 1.0f);
  float4 v = ((const float4*)(h + (size_t)s * DD))[lane];
  float* o = agg + ((size_t)r * NN + (size_t)d) * DD + lane * 4;
  atomAddF(o + 0, v.x * sc);
  atomAddF(o + 1, v.y * sc);
  atomAddF(o + 2, v.z * sc);
  atomAddF(o + 3, v.w * sc);
}

// ---------------------------------------------- dense transform via fp32 WMMA
// 32 rows x 128 cols per block: out = agg0@W[0] + agg1@W[1] + h@root + bias
// A tiles staged to LDS once (async); wave w owns col tile [16w,16w+16),
// two row tiles per wave (B-fragment reuse).
__global__ void __launch_bounds__(256) k_transform(
    const float* __restrict__ agg, const float* __restrict__ h,
    const float* __restrict__ W, const float* __restrict__ root,
    const float* __restrict__ bias, float* __restrict__ out) {
  __shared__ float As[3][32][132];  // +4 pad -> conflict-free fragment reads
  const int t = threadIdx.x;
  const int row0 = blockIdx.x * 32;

  const float* Ag[3] = {agg, agg + (size_t)NN * DD, h};
  const float* Bp[3] = {W, W + DD * DD, root};

  // stage A tiles: 3 * 32 * 128 floats = 3072 float4, 12 per thread
#pragma unroll
  for (int i = 0; i < 12; ++i) {
    int f = t + i * 256;
    int m = f >> 10;
    int rem = f & 1023;
    int row = rem >> 5;
    int c4 = (rem & 31) << 2;
    async_copy16(&As[m][row][c4], Ag[m] + (size_t)(row0 + row) * DD + c4);
  }
  async_wait0();
  __syncthreads();

  const int lane = t & 31;
  const int wv = t >> 5;
  const int l15 = lane & 15;
  const int koff = 2 * (lane >> 4);   // ISA A/B lane K offset
  const int bcol = wv * 16 + l15;
  const int rbase = 8 * (lane >> 4);

  v8f acc0 = {0.f, 0.f, 0.f, 0.f, 0.f, 0.f, 0.f, 0.f};
  v8f acc1 = acc0;
  for (int m = 0; m < 3; ++m) {
    const float* B = Bp[m] + bcol;
#pragma unroll 8
    for (int k = 0; k < DD; k += 4) {
      v2f b;
      b.x = B[(size_t)(k + koff) * DD];
      b.y = B[(size_t)(k + koff + 1) * DD];
      v2f a0 = *(const v2f*)&As[m][l15][k + koff];
      v2f a1 = *(const v2f*)&As[m][16 + l15][k + koff];
      acc0 = wmma_f32(a0, b, acc0);
      acc1 = wmma_f32(a1, b, acc1);
    }
  }
  float bv = bias[bcol];
  int r0 = row0 + rbase;
#pragma unroll
  for (int v = 0; v < 8; ++v) {
    out[(size_t)(r0 + v) * DD + bcol] = acc0[v] + bv;
    out[(size_t)(r0 + 16 + v) * DD + bcol] = acc1[v] + bv;
  }
}

// ----------------------------------------------------- BN stats (1 block/column)
__global__ void k_stats(const float* __restrict__ pre, const float* __restrict__ g,
                        const float* __restrict__ b, float* __restrict__ scale,
                        float* __restrict__ shift) {
  __shared__ float ss[256], ss2[256];
  const int col = blockIdx.x, t = threadIdx.x;
  float s = 0.f, s2 = 0.f;
  for (int i = t; i < NN; i += 256) {
    float v = pre[(size_t)i * DD + col];
    s += v;
    s2 += v * v;
  }
  ss[t] = s;
  ss2[t] = s2;
  __syncthreads();
  for (int st = 128; st > 0; st >>= 1) {
    if (t < st) {
      ss[t] += ss[t + st];
      ss2[t] += ss2[t + st];
    }
    __syncthreads();
  }
  if (t == 0) {
    float mu = ss[0] / (float)NN;
    float var = ss2[0] / (float)NN - mu * mu;
    float sc = rsqrtf(var + BN_EPS) * g[col];
    scale[col] = sc;
    shift[col] = b[col] - mu * sc;
  }
}

// ------------------------------------------------------------ BN + ReLU (vec4)
__global__ void k_bnrelu(const float* __restrict__ in, float* __restrict__ out,
                         const float* __restrict__ scale,
                         const float* __restrict__ shift) {
  int i = blockIdx.x * blockDim.x + threadIdx.x;  // float4 index
  if (i >= NN * DD / 4) return;
  int c4 = (i * 4) & (DD - 1);
  float4 v = ((const float4*)in)[i];
  float4 sc = *(const float4*)(scale + c4);
  float4 sh = *(const float4*)(shift + c4);
  v.x = fmaxf(v.x * sc.x + sh.x, 0.f);
  v.y = fmaxf(v.y * sc.y + sh.y, 0.f);
  v.z = fmaxf(v.z * sc.z + sh.z, 0.f);
  v.w = fmaxf(v.w * sc.w + sh.w, 0.f);
  ((float4*)out)[i] = v;
}

// --------------------------------------- fused 3-layer edge decoder (32 edges/blk)
// feat = [z[src], z[dst]] (32x256) -> relu(@dW1) 32x128 -> relu(@dW2) 32x64 -> @dW3
__global__ void __launch_bounds__(256) k_decoder(
    const float* __restrict__ z, const int* __restrict__ ei,
    const float* __restrict__ dW1, const float* __restrict__ db1,
    const float* __restrict__ dW2, const float* __restrict__ db2,
    const float* __restrict__ dW3, const float* __restrict__ db3,
    float* __restrict__ out) {
  __shared__ float feat[32][260];  // +4 pad: conflict-free A-fragment reads
  __shared__ float h1s[32][132];
  __shared__ float h2s[32][68];
  const int t = threadIdx.x;
  const int e0 = blockIdx.x * 32;

  {  // gather feat via async copies: thread t -> row t>>3, 32 cols at (t&7)*32
    int row = t >> 3;
    int cseg = t & 7;
    int node = (cseg < 4) ? ei[e0 + row] : ei[EE + e0 + row];
    const float* gsrc =
        z + (size_t)node * DD + ((cseg < 4) ? cseg * 32 : cseg * 32 - 128);
    float* ldst = &feat[row][cseg * 32];
#pragma unroll
    for (int i = 0; i < 8; ++i) async_copy16(ldst + i * 4, gsrc + i * 4);
  }
  async_wait0();
  __syncthreads();

  const int lane = t & 31;
  const int wv = t >> 5;
  const int koff = 2 * (lane >> 4);
  const int l15 = lane & 15;
  const int rbase = 8 * (lane >> 4);

  {  // stage 1: h1 = relu(feat @ dW1 + db1); wave w -> cols [16w,16w+16), 2 row tiles
    int bcol = wv * 16 + l15;
    v8f acc0 = {0.f, 0.f, 0.f, 0.f, 0.f, 0.f, 0.f, 0.f};
    v8f acc1 = acc0;
#pragma unroll 8
    for (int k = 0; k < 2 * DD; k += 4) {
      v2f b;
      b.x = dW1[(size_t)(k + koff) * DD + bcol];
      b.y = dW1[(size_t)(k + koff + 1) * DD + bcol];
      v2f a0 = *(const v2f*)&feat[l15][k + koff];
      v2f a1 = *(const v2f*)&feat[16 + l15][k + koff];
      acc0 = wmma_f32(a0, b, acc0);
      acc1 = wmma_f32(a1, b, acc1);
    }
    float bv = db1[bcol];
#pragma unroll
    for (int v = 0; v < 8; ++v) {
      h1s[rbase + v][bcol] = fmaxf(acc0[v] + bv, 0.f);
      h1s[16 + rbase + v][bcol] = fmaxf(acc1[v] + bv, 0.f);
    }
  }
  __syncthreads();

  {  // stage 2: h2 = relu(h1 @ dW2 + db2), 32x64; wave -> (rowtile wv>>2, coltile wv&3)
    int rt = (wv >> 2) * 16;
    int bcol = (wv & 3) * 16 + l15;
    v8f acc = {0.f, 0.f, 0.f, 0.f, 0.f, 0.f, 0.f, 0.f};
#pragma unroll 8
    for (int k = 0; k < DD; k += 4) {
      v2f b;
      b.x = dW2[(size_t)(k + koff) * 64 + bcol];
      b.y = dW2[(size_t)(k + koff + 1) * 64 + bcol];
      v2f a = *(const v2f*)&h1s[rt + l15][k + koff];
      acc = wmma_f32(a, b, acc);
    }
    float bv = db2[bcol];
#pragma unroll
    for (int v = 0; v < 8; ++v)
      h2s[rt + rbase + v][bcol] = fmaxf(acc[v] + bv, 0.f);
  }
  __syncthreads();

  if (t < 32) {  // stage 3: logits
    float s = db3[0];
#pragma unroll 8
    for (int j = 0; j < 64; ++j) s += h2s[t][j] * dW3[j];
    out[e0 + t] = s;
  }
}

extern "C" void kernel_launch(void* const* d_in, const int* in_sizes, int n_in,
                              void* d_out, int out_size, void* d_ws, size_t ws_size,
                              hipStream_t stream) {
  (void)in_sizes; (void)n_in; (void)out_size; (void)ws_size;
  const int* ei = (const int*)d_in[0];
  const int* et = (const int*)d_in[1];
  const float* emb = (const float*)d_in[2];
  const float* W0 = (const float*)d_in[3];
  const float* root0 = (const float*)d_in[4];
  const float* bias0 = (const float*)d_in[5];
  const float* g0 = (const float*)d_in[6];
  const float* b0 = (const float*)d_in[7];
  const float* W1 = (const float*)d_in[8];
  const float* root1 = (const float*)d_in[9];
  const float* bias1 = (const float*)d_in[10];
  const float* g1 = (const float*)d_in[11];
  const float* b1 = (const float*)d_in[12];
  const float* dW1 = (const float*)d_in[13];
  const float* db1 = (const float*)d_in[14];
  const float* dW2 = (const float*)d_in[15];
  const float* db2 = (const float*)d_in[16];
  const float* dW3 = (const float*)d_in[17];
  const float* db3 = (const float*)d_in[18];
  float* out = (float*)d_out;

  float* cnt = (float*)d_ws;                       // N*R
  float* agg = cnt + (size_t)NN * RR;              // R*N*D
  float* pre = agg + (size_t)RR * NN * DD;         // N*D (becomes z in-place)
  float* h = pre + (size_t)NN * DD;                // N*D
  float* scale = h + (size_t)NN * DD;              // D
  float* shift = scale + DD;                       // D

  (void)hipMemsetAsync(cnt, 0, (size_t)NN * RR * sizeof(float), stream);
  k_count<<<EE / 256, 256, 0, stream>>>(ei, et, cnt);

  // ---- layer 0
  (void)hipMemsetAsync(agg, 0, (size_t)RR * NN * DD * sizeof(float), stream);
  k_aggregate<<<EE / 8, 256, 0, stream>>>(emb, ei, et, cnt, agg);
  k_transform<<<NN / 32, 256, 0, stream>>>(agg, emb, W0, root0, bias0, pre);
  k_stats<<<DD, 256, 0, stream>>>(pre, g0, b0, scale, shift);
  k_bnrelu<<<NN * DD / 4 / 256, 256, 0, stream>>>(pre, h, scale, shift);

  // ---- layer 1
  (void)hipMemsetAsync(agg, 0, (size_t)RR * NN * DD * sizeof(float), stream);
  k_aggregate<<<EE / 8, 256, 0, stream>>>(h, ei, et, cnt, agg);
  k_transform<<<NN / 32, 256, 0, stream>>>(agg, h, W1, root1, bias1, pre);
  k_stats<<<DD, 256, 0, stream>>>(pre, g1, b1, scale, shift);
  k_bnrelu<<<NN * DD / 4 / 256, 256, 0, stream>>>(pre, pre, scale, shift);  // z

  // ---- fused edge decoder
  k_decoder<<<EE / 32, 256, 0, stream>>>(pre, ei, dW1, db1, dW2, db2, dW3, db3,
                                         out);
}